// GraphConvAutoencoder_82085414961635
// MI455X (gfx1250) — compile-verified
//
#include <hip/hip_runtime.h>
#include <math.h>

// ---------------------------------------------------------------------------
// GraphConv autoencoder for MI455X (gfx1250).
// Dense GEMMs use V_WMMA_F32_16X16X4_F32 (full fp32, matches reference
// precision).  Edge aggregation uses native global_atomic_add_f32
// (unsafeAtomicAdd) -- memory-bound path, ~3.9GB irregular traffic total.
// ---------------------------------------------------------------------------

typedef __attribute__((ext_vector_type(2))) float v2f;
typedef __attribute__((ext_vector_type(8))) float v8f;

// -------------------------- WMMA GEMM --------------------------------------
// D[rows, dout] (+)= A[rows, ldA(first K cols)] @ B[K, dout]
// One wave per 16x16 output tile.  8 waves (256 threads) per block.
// A fragment (16x4 f32, ISA 7.12.2): lanes 0-15 row m K={k0,k0+1},
//                                    lanes 16-31 row m K={k0+2,k0+3}.
// B fragment (4x16): vgpr0 = row k0 (lanes0-15) / row k0+2 (lanes16-31),
//                    vgpr1 = row k0+1 / k0+3.
// C/D (16x16 f32): vgpr j -> row j (lanes0-15), row j+8 (lanes16-31).
__global__ __launch_bounds__(256) void gemm_wmma_f32(
    const float* __restrict__ A, const float* __restrict__ B,
    float* __restrict__ D, const float* __restrict__ bias,
    int rows, int K, int ldA, int dout, int acc, int act)
{
    const int wave = threadIdx.x >> 5;
    const int lane = threadIdx.x & 31;
    const int tileRow = blockIdx.x * 8 + wave;
    if (tileRow * 16 >= rows) return;              // uniform per wave
    const int col0 = blockIdx.y << 4;
    const int hi = lane >> 4;                      // 0 or 1
    const int m  = lane & 15;

    const float* __restrict__ arow = A + (size_t)(tileRow * 16 + m) * ldA;

    v8f c;
    const int rbase = tileRow * 16 + hi * 8;
    if (acc) {
#pragma unroll
        for (int j = 0; j < 8; ++j)
            c[j] = D[(size_t)(rbase + j) * dout + col0 + m];
    } else {
#pragma unroll
        for (int j = 0; j < 8; ++j) c[j] = 0.0f;
    }

    for (int k0 = 0; k0 < K; k0 += 4) {
        const int ka = k0 + 2 * hi;
        v2f a, b;
        a[0] = arow[ka];
        a[1] = arow[ka + 1];
        b[0] = B[(size_t)ka * dout + col0 + m];
        b[1] = B[(size_t)(ka + 1) * dout + col0 + m];
        c = __builtin_amdgcn_wmma_f32_16x16x4_f32(
                false, a, false, b, (short)0, c, false, false);
    }

    const float bv = bias ? bias[col0 + m] : 0.0f;
#pragma unroll
    for (int j = 0; j < 8; ++j) {
        float v = c[j] + bv;
        if (act == 1)      v = v > 0.0f ? v : 0.0f;
        else if (act == 2) v = tanhf(v);
        D[(size_t)(rbase + j) * dout + col0 + m] = v;
    }
}

// -------------------------- helpers ----------------------------------------
__global__ __launch_bounds__(256) void fill_f32(float* p, float v, int n)
{
    int i = blockIdx.x * 256 + threadIdx.x;
    if (i < n) p[i] = v;
}

__global__ __launch_bounds__(256) void count_deg(const int* __restrict__ dst,
                                                 float* __restrict__ deg, int E)
{
    int e = blockIdx.x * 256 + threadIdx.x;
    if (e < E) unsafeAtomicAdd(&deg[dst[e]], 1.0f);
}

__global__ __launch_bounds__(256) void make_dinv(const float* __restrict__ deg,
                                                 float* __restrict__ dinv, int N)
{
    int i = blockIdx.x * 256 + threadIdx.x;
    if (i < N) {
        float d = deg[i];
        dinv[i] = d > 0.0f ? rsqrtf(d) : 0.0f;
    }
}

__global__ __launch_bounds__(256) void make_norm(const int* __restrict__ src,
                                                 const int* __restrict__ dst,
                                                 const float* __restrict__ dinv,
                                                 float* __restrict__ norm, int E)
{
    int e = blockIdx.x * 256 + threadIdx.x;
    if (e < E) norm[e] = dinv[src[e]] * dinv[dst[e]];
}

// agg[i,:] = h[i,:] * dinv[i]^2   (self-loop contribution; also inits agg)
__global__ __launch_bounds__(256) void selfloop_init(const float* __restrict__ h,
                                                     const float* __restrict__ dinv,
                                                     float* __restrict__ agg,
                                                     long total, int dshift)
{
    long t = (long)blockIdx.x * 256 + threadIdx.x;
    if (t >= total) return;
    long i = t >> dshift;
    float di = dinv[i];
    agg[t] = h[t] * di * di;
}

// agg[dst,:] += norm[e] * h[src,:]; one thread per (edge, 4-feature chunk)
__global__ __launch_bounds__(256) void edge_agg(const int* __restrict__ src,
                                                const int* __restrict__ dst,
                                                const float* __restrict__ norm,
                                                const float* __restrict__ h,
                                                float* __restrict__ agg,
                                                long total, int chunkShift, int d)
{
    long t = (long)blockIdx.x * 256 + threadIdx.x;
    if (t >= total) return;
    long e = t >> chunkShift;
    int  q = (int)(t & ((1 << chunkShift) - 1));
    int  si = src[e], di = dst[e];
    float w = norm[e];
    const float4 v = *((const float4*)(h + (size_t)si * d) + q);
    float* ap = agg + (size_t)di * d + q * 4;
    unsafeAtomicAdd(ap + 0, v.x * w);
    unsafeAtomicAdd(ap + 1, v.y * w);
    unsafeAtomicAdd(ap + 2, v.z * w);
    unsafeAtomicAdd(ap + 3, v.w * w);
}

// out = act(agg + bias)
__global__ __launch_bounds__(256) void bias_act(const float* __restrict__ agg,
                                                const float* __restrict__ b,
                                                float* __restrict__ out,
                                                long total, int d, int act)
{
    long t = (long)blockIdx.x * 256 + threadIdx.x;
    if (t >= total) return;
    float v = agg[t] + b[(int)(t & (d - 1))];
    if (act == 1)      v = v > 0.0f ? v : 0.0f;
    else if (act == 2) v = tanhf(v);
    out[t] = v;
}

// ---------------------------------------------------------------------------
static inline int cdiv(long a, long b) { return (int)((a + b - 1) / b); }

extern "C" void kernel_launch(void* const* d_in, const int* in_sizes, int n_in,
                              void* d_out, int out_size, void* d_ws, size_t ws_size,
                              hipStream_t stream)
{
    const float* x   = (const float*)d_in[0];
    const int*   ei  = (const int*)d_in[1];
    const float* W1  = (const float*)d_in[2];   const float* b1  = (const float*)d_in[3];
    const float* W2  = (const float*)d_in[4];   const float* b2  = (const float*)d_in[5];
    const float* W3  = (const float*)d_in[6];   const float* b3  = (const float*)d_in[7];
    const float* We  = (const float*)d_in[8];   const float* be  = (const float*)d_in[9];
    const float* Wd  = (const float*)d_in[10];  const float* bd  = (const float*)d_in[11];
    const float* dW1 = (const float*)d_in[12];  const float* db1 = (const float*)d_in[13];
    const float* dW2 = (const float*)d_in[14];  const float* db2 = (const float*)d_in[15];
    const float* dW3 = (const float*)d_in[16];  const float* db3 = (const float*)d_in[17];

    const int N = in_sizes[0] / 128;
    const int E = in_sizes[1] / 2;
    const int* src = ei;
    const int* dst = ei + E;

    // ---- workspace layout (floats) ----
    float* ws   = (float*)d_ws;
    float* deg  = ws;                      // N
    float* dinv = deg  + N;                // N
    float* nrm  = dinv + N;                // E
    float* bufA = nrm  + E;                // N*128 (pre-aggregation h)
    float* bufB = bufA + (size_t)N * 128;  // N*128 (aggregation accumulator / z)
    float* x1b  = bufB + (size_t)N * 128;  // N*16
    float* x2b  = x1b  + (size_t)N * 16;   // N*32
    float* x3b  = x2b  + (size_t)N * 32;   // N*64  (later reused for decoder d)

    // ---- normalization ----
    fill_f32 <<<cdiv(N, 256), 256, 0, stream>>>(deg, 1.0f, N);   // self loops
    count_deg<<<cdiv(E, 256), 256, 0, stream>>>(dst, deg, E);
    make_dinv<<<cdiv(N, 256), 256, 0, stream>>>(deg, dinv, N);
    make_norm<<<cdiv(E, 256), 256, 0, stream>>>(src, dst, dinv, nrm, E);

    auto gemm = [&](const float* A, const float* B, float* D, const float* bias,
                    int K, int ldA, int dout, int acc, int act) {
        dim3 grid(cdiv((N + 15) / 16, 8), dout / 16);
        gemm_wmma_f32<<<grid, 256, 0, stream>>>(A, B, D, bias, N, K, ldA, dout, acc, act);
    };

    // one GCN layer: out = act( scatter(norm * (A@W)[src] -> dst) + selfloop + b )
    auto gcn = [&](const float* A, int din, const float* W,
                   const float* A2, int din2, const float* Wpart2,
                   const float* bias, int dout, float* out, int act) {
        gemm(A, W, bufA, nullptr, din, din, dout, 0, 0);
        if (A2) gemm(A2, Wpart2, bufA, nullptr, din2, din2, dout, 1, 0);
        const long nt = (long)N * dout;
        const int dshift = __builtin_ctz(dout);
        selfloop_init<<<cdiv(nt, 256), 256, 0, stream>>>(bufA, dinv, bufB, nt, dshift);
        const int cshift = __builtin_ctz(dout >> 2);        // chunks = dout/4
        const long et = (long)E << cshift;
        edge_agg<<<cdiv(et, 256), 256, 0, stream>>>(src, dst, nrm, bufA, bufB,
                                                    et, cshift, dout);
        bias_act<<<cdiv(nt, 256), 256, 0, stream>>>(bufB, bias, out, nt, dout, act);
    };

    // ---- encoder ----
    gcn(x,   128, W1, nullptr, 0, nullptr, b1, 16, x1b, 1);   // x1 = relu(gcn)
    gcn(x1b,  16, W2, nullptr, 0, nullptr, b2, 32, x2b, 1);   // x2
    gcn(x2b,  32, W3, nullptr, 0, nullptr, b3, 64, x3b, 1);   // x3

    // ---- bottleneck (dense, fused bias/act in GEMM epilogue) ----
    gemm(x3b, We, bufB, be, 64, 64, 32, 0, 0);                // z  [N,32] -> bufB
    gemm(bufB, Wd, x3b, bd, 32, 32, 64, 0, 1);                // d  [N,64] -> x3b (x3 dead)

    // ---- decoder (skip concats done as two accumulating GEMMs) ----
    gcn(x3b, 64, dW1, x2b, 32, dW1 + 64 * 32, db1, 32, x2b, 1);  // [d | x2] @ dW1
    gcn(x2b, 32, dW2, x1b, 16, dW2 + 32 * 16, db2, 16, x1b, 1);  // [d | x1] @ dW2
    gcn(x1b, 16, dW3, nullptr, 0, nullptr, db3, 128, (float*)d_out, 2); // tanh
}